// ProposalLayer_25580825215447
// MI455X (gfx1250) — compile-verified
//
#include <hip/hip_runtime.h>
#include <stdint.h>

// ---------------- problem constants (from reference) ----------------
#define BATCH        16
#define NUM_ANCHORS  261888
#define K_SEL        6000      // PRE_NMS_LIMIT (== k, since N > 6000)
#define K_PAD        8192      // bitonic sort size (power of 2 >= K_SEL + slack)
#define BOX_PAD      6144      // padded box count per batch (multiple of 1024)
#define OUT_N        1000      // PROPOSAL_COUNT
#define NMS_THR      0.7f

// order-preserving float -> uint key (descending float == descending uint)
__device__ __forceinline__ uint32_t order_key(float f) {
    uint32_t u = __float_as_uint(f);
    return (u & 0x80000000u) ? ~u : (u | 0x80000000u);
}

__device__ __forceinline__ float clamp01(float v) {
    return fminf(fmaxf(v, 0.0f), 1.0f);
}

// Block-cooperative: find bin B (descending order) such that
//   base + count(bins > B) < K_SEL <= base + count(bins >= B)
// Hierarchical: 32-bin group sums in parallel, then thread 0 scans
// <= ngroups + 32 entries serially (instead of nbins).
__device__ __forceinline__ void find_threshold(const unsigned* hist, int nbins,
                                               unsigned base, unsigned* s_grp,
                                               unsigned* outBin, unsigned* outAbove) {
    const int tid = threadIdx.x;
    const int ngroups = nbins >> 5;
    if (tid < ngroups) {
        unsigned s = 0;
        #pragma unroll 8
        for (int j = 0; j < 32; ++j) s += hist[(tid << 5) | j];
        s_grp[tid] = s;
    }
    __syncthreads();
    if (tid == 0) {
        unsigned cum = base;
        int g = ngroups - 1;
        for (; g > 0; --g) {
            unsigned c = s_grp[g];
            if (cum + c >= K_SEL) break;
            cum += c;
        }
        int bin = (g << 5) | 31;
        for (; bin > (g << 5); --bin) {
            unsigned c = hist[bin];
            if (cum + c >= K_SEL) break;
            cum += c;
        }
        *outBin = (unsigned)bin;
        *outAbove = cum;
    }
    __syncthreads();
}

// =====================================================================
// Kernel 1: exact top-6000 (3-level radix-refined threshold + LDS bitonic
// sort), fused with box decode. One workgroup (32 wave32) per batch.
// Dynamic LDS: u64 sbuf[8192] (64KB); first 2048 u32 alias as histogram.
// =====================================================================
__global__ __launch_bounds__(1024) void topk_decode_kernel(
    const float* __restrict__ probs,    // [B, N, 2]
    const float* __restrict__ bbox,     // [B, N, 4]
    const float* __restrict__ anchors,  // [B, N, 4]
    float4* __restrict__ ws_boxes)      // [B, BOX_PAD] decoded clipped boxes
{
    extern __shared__ unsigned char smem_raw[];
    unsigned long long* sbuf = (unsigned long long*)smem_raw;  // [K_PAD]
    unsigned* hist = (unsigned*)smem_raw;                      // alias [2048]
    __shared__ unsigned s_grp[64];
    __shared__ unsigned s_ctrl[8];

    const int b    = blockIdx.x;
    const int tid  = threadIdx.x;
    const int lane = tid & 31;
    const float* sc = probs + (size_t)b * NUM_ANCHORS * 2 + 1;  // fg prob, stride 2

    // ---- level 1 histogram: key bits [31:21] (2048 bins) ----
    for (int i = tid; i < 2048; i += 1024) hist[i] = 0;
    __syncthreads();
    for (int i = tid; i < NUM_ANCHORS; i += 1024) {
        __builtin_prefetch(&sc[(size_t)(i + 4096) * 2], 0, 3);  // global_prefetch_b8, near
        uint32_t key = order_key(sc[(size_t)i * 2]);
        atomicAdd(&hist[key >> 21], 1u);
    }
    __syncthreads();
    find_threshold(hist, 2048, 0u, s_grp, &s_ctrl[0], &s_ctrl[1]);
    const unsigned B1 = s_ctrl[0], above1 = s_ctrl[1];
    __syncthreads();

    // ---- level 2: key bits [20:10] within bin B1 ----
    for (int i = tid; i < 2048; i += 1024) hist[i] = 0;
    __syncthreads();
    for (int i = tid; i < NUM_ANCHORS; i += 1024) {
        uint32_t key = order_key(sc[(size_t)i * 2]);
        if ((key >> 21) == B1) atomicAdd(&hist[(key >> 10) & 0x7FFu], 1u);
    }
    __syncthreads();
    find_threshold(hist, 2048, above1, s_grp, &s_ctrl[2], &s_ctrl[3]);
    const unsigned B2 = s_ctrl[2], above2 = s_ctrl[3];
    __syncthreads();

    // ---- level 3: key bits [9:0] within (B1,B2) -> exact K-th key ----
    for (int i = tid; i < 2048; i += 1024) hist[i] = 0;
    __syncthreads();
    const unsigned hi22 = (B1 << 11) | B2;
    for (int i = tid; i < NUM_ANCHORS; i += 1024) {
        uint32_t key = order_key(sc[(size_t)i * 2]);
        if ((key >> 10) == hi22) atomicAdd(&hist[key & 0x3FFu], 1u);
    }
    __syncthreads();
    find_threshold(hist, 1024, above2, s_grp, &s_ctrl[4], &s_ctrl[5]);
    const unsigned Kfull = (hi22 << 10) | s_ctrl[4];
    if (tid == 0) s_ctrl[6] = 0;     // candidate counter
    __syncthreads();

    // ---- gather candidates (key >= Kfull); wave-aggregated LDS atomics ----
    for (int i = tid; i < NUM_ANCHORS; i += 1024) {
        uint32_t key = order_key(sc[(size_t)i * 2]);
        bool cand = (key >= Kfull);
        unsigned m32 = (unsigned)__ballot(cand);
        unsigned wbase = 0;
        if (lane == 0 && m32) wbase = atomicAdd(&s_ctrl[6], (unsigned)__popc(m32));
        wbase = (unsigned)__shfl((int)wbase, 0, 32);
        if (cand) {
            unsigned pos = wbase + (unsigned)__popc(m32 & ((1u << lane) - 1u));
            if (pos < K_PAD)
                sbuf[pos] = ((unsigned long long)key << 32) |
                            (unsigned long long)(0xFFFFFFFFu - (unsigned)i);
        }
    }
    __syncthreads();
    unsigned cnt = s_ctrl[6]; if (cnt > K_PAD) cnt = K_PAD;
    for (int i = (int)cnt + tid; i < K_PAD; i += 1024) sbuf[i] = 0ull;
    __syncthreads();

    // ---- bitonic sort, descending on (key, ~idx): keys desc, idx asc on ties ----
    for (int k = 2; k <= K_PAD; k <<= 1) {
        for (int j = k >> 1; j > 0; j >>= 1) {
            for (int i = tid; i < K_PAD; i += 1024) {
                int ixj = i ^ j;
                if (ixj > i) {
                    unsigned long long a = sbuf[i], c = sbuf[ixj];
                    bool up = ((i & k) == 0);
                    if (up ? (a < c) : (a > c)) { sbuf[i] = c; sbuf[ixj] = a; }
                }
            }
            __syncthreads();
        }
    }

    // ---- decode top-6000 boxes (gather anchors/deltas, apply, clip) ----
    const float4* anc4 = (const float4*)(anchors + (size_t)b * NUM_ANCHORS * 4);
    const float4* dlt4 = (const float4*)(bbox    + (size_t)b * NUM_ANCHORS * 4);
    for (int i = tid; i < BOX_PAD; i += 1024) {
        float4 o = make_float4(0.f, 0.f, 0.f, 0.f);
        if (i < K_SEL) {
            unsigned idx = 0xFFFFFFFFu - (unsigned)(sbuf[i] & 0xFFFFFFFFull);
            float4 a = anc4[idx];            // [y1, x1, y2, x2]
            float4 d = dlt4[idx];
            d.x *= 0.1f; d.y *= 0.1f; d.z *= 0.2f; d.w *= 0.2f;  // RPN_BBOX_STD_DEV
            float h  = a.z - a.x;
            float w  = a.w - a.y;
            float cy = a.x + 0.5f * h + d.x * h;
            float cx = a.y + 0.5f * w + d.y * w;
            h *= expf(d.z);
            w *= expf(d.w);
            float y1 = cy - 0.5f * h;
            float x1 = cx - 0.5f * w;
            o = make_float4(clamp01(y1), clamp01(x1), clamp01(y1 + h), clamp01(x1 + w));
        }
        ws_boxes[(size_t)b * BOX_PAD + i] = o;
    }
}

// =====================================================================
// Kernel 2: greedy NMS per batch. Boxes staged to LDS with the CDNA5
// async-to-LDS path (96KB boxes resident in the 320KB WGP LDS), 6000-bit
// validity bitmap in LDS with ds_and atomics.
// =====================================================================
__global__ __launch_bounds__(1024) void nms_kernel(
    const float4* __restrict__ ws_boxes,  // [B, BOX_PAD]
    float* __restrict__ out)              // [B, OUT_N, 4]
{
    extern __shared__ unsigned char smem_raw[];
    float4*   sbox  = (float4*)smem_raw;                              // [BOX_PAD]
    unsigned* valid = (unsigned*)(smem_raw + (size_t)BOX_PAD * 16);   // [192]
    __shared__ unsigned s_ctrl[4];  // 0: bitmap search word, 1: emit count, 2: pivot

    const int b   = blockIdx.x;
    const int tid = threadIdx.x;
    const float4* src = ws_boxes + (size_t)b * BOX_PAD;
    float* ob = out + (size_t)b * OUT_N * 4;

    // ---- async global -> LDS stage of all 6144 boxes (b128 per lane) ----
    for (int i = tid; i < BOX_PAD; i += 1024) {
        unsigned lds_off = (unsigned)(size_t)(const void*)&sbox[i];
        unsigned long long gaddr = (unsigned long long)(size_t)(const void*)&src[i];
        asm volatile("global_load_async_to_lds_b128 %0, %1, off"
                     :: "v"(lds_off), "v"(gaddr) : "memory");
    }
    // overlap with the async copies: zero output, init 6000-bit validity bitmap
    for (int i = tid; i < OUT_N * 4; i += 1024) ob[i] = 0.0f;
    if (tid < 192)
        valid[tid] = (tid < 187) ? 0xFFFFFFFFu : (tid == 187 ? 0x0000FFFFu : 0u);
    if (tid == 0) { s_ctrl[0] = 0; s_ctrl[1] = 0; }
    asm volatile("s_wait_asynccnt 0" ::: "memory");
    __syncthreads();

    // ---- greedy NMS loop: emit next survivor, suppress IoU > 0.7 ----
    while (true) {
        if (tid == 0) {
            int pivot = -1;
            unsigned w = s_ctrl[0];
            if (s_ctrl[1] < OUT_N) {
                while (w < 192) {
                    unsigned m = valid[w];
                    if (m) { pivot = (int)(w * 32u + (unsigned)(__ffs(m) - 1)); break; }
                    ++w;
                }
            }
            s_ctrl[0] = w;
            if (pivot >= 0) {
                valid[w] &= ~(1u << (pivot & 31));   // self-suppress
                unsigned e = s_ctrl[1]++;
                s_ctrl[2] = (unsigned)pivot;
                float4 p = sbox[pivot];
                *(float4*)(ob + (size_t)e * 4) = p;  // global_store_b128
            } else {
                s_ctrl[2] = 0xFFFFFFFFu;
            }
        }
        __syncthreads();
        const unsigned pv = s_ctrl[2];
        if (pv == 0xFFFFFFFFu) break;

        const float4 p = sbox[pv];
        const float parea = (p.z - p.x) * (p.w - p.y);
        // only boxes after the pivot can still be valid (pivot = first set bit)
        for (int i = (int)pv + 1 + tid; i < K_SEL; i += 1024) {
            if (!((valid[i >> 5] >> (i & 31)) & 1u)) continue;
            float4 q = sbox[i];
            float iy1 = fmaxf(p.x, q.x), ix1 = fmaxf(p.y, q.y);
            float iy2 = fminf(p.z, q.z), ix2 = fminf(p.w, q.w);
            float inter = fmaxf(iy2 - iy1, 0.0f) * fmaxf(ix2 - ix1, 0.0f);
            float qarea = (q.z - q.x) * (q.w - q.y);
            float iou = inter / fmaxf(parea + qarea - inter, 1e-12f);
            if (iou > NMS_THR) atomicAnd(&valid[i >> 5], ~(1u << (i & 31)));
        }
        __syncthreads();
    }
}

// =====================================================================
extern "C" void kernel_launch(void* const* d_in, const int* in_sizes, int n_in,
                              void* d_out, int out_size, void* d_ws, size_t ws_size,
                              hipStream_t stream) {
    const float* probs   = (const float*)d_in[0];  // [16, 261888, 2]
    const float* bbox    = (const float*)d_in[1];  // [16, 261888, 4]
    const float* anchors = (const float*)d_in[2];  // [16, 261888, 4]
    float* out = (float*)d_out;                    // [16, 1000, 4]
    float4* ws_boxes = (float4*)d_ws;              // needs 16*6144*16 = 1.5 MB

    const size_t smem1 = (size_t)K_PAD * sizeof(unsigned long long);      // 64 KB
    const size_t smem2 = (size_t)BOX_PAD * sizeof(float4) + 192 * 4;      // ~96.8 KB

    topk_decode_kernel<<<BATCH, 1024, smem1, stream>>>(probs, bbox, anchors, ws_boxes);
    nms_kernel<<<BATCH, 1024, smem2, stream>>>(ws_boxes, out);
}